// VONet_8529805050033
// MI455X (gfx1250) — compile-verified
//
#include <hip/hip_runtime.h>
#include <hip/hip_bf16.h>
#include <math.h>

typedef __attribute__((ext_vector_type(2))) float v2f;
typedef __attribute__((ext_vector_type(8))) float v8f;

#define DIMV   384
#define EDGES  65536
#define NSEG   4096
#define CDIMK  882

#define ACT_NONE    0
#define ACT_RELU    1
#define ACT_SIGMOID 2

// ---------------------------------------------------------------------------
// Generic WMMA GEMM:  Y[M,N] (+)= act(X[M,K] @ Wm[K,N] + bias[N])
// 256 threads = 8 waves; block tile 32(M) x 64(N); K in 32-chunks staged via
// double-buffered LDS. Software pipeline per chunk:
//   global_load(next chunk -> regs) ; 8x v_wmma_f32_16x16x4_f32 on cur LDS buf ;
//   ds_store(regs -> other buf) ; barrier.
// The s_wait_loadcnt for the staging stores lands AFTER the WMMAs, so global
// latency is hidden behind the matrix pipe. KAL=true (K%32==0, 16B-aligned
// rows) uses unguarded float4 loads; KAL=false handles ragged K (corr, K=882)
// by zero-padding in LDS.
// ---------------------------------------------------------------------------
template<int ACT, bool ACCUM, bool KAL>
__global__ __launch_bounds__(256)
void gemm_wmma(const float* __restrict__ X, const float* __restrict__ Wm,
               const float* __restrict__ bias, float* __restrict__ Y,
               int M, int K, int N)
{
    __shared__ __align__(16) float Xs[2][32][36];   // pad -> conflict-free col reads
    __shared__ __align__(16) float Ws[2][32][80];   // pad -> half-wave bank split

    const int tid   = threadIdx.x;
    const int lane  = tid & 31;
    const int wave  = tid >> 5;
    const int waveM = wave >> 2;        // 0..1
    const int waveN = wave & 3;         // 0..3
    const int m0    = blockIdx.x * 32;
    const int n0    = blockIdx.y * 64;

    const int hi  = lane >> 4;          // 0/1 half-wave
    const int l15 = lane & 15;
    const int kb2 = hi * 2;
    const int mrow = waveM * 16 + l15;
    const int ncol = waveN * 16 + l15;

    // per-thread staging coordinates
    const int xrow = tid >> 3;               // 0..31
    const int xc4  = (tid & 7) << 2;         // 0..28
    const int wr0  = tid >> 4;               // 0..15 (second row = +16)
    const int wc4  = (tid & 15) << 2;        // 0..60
    const float* xbase = X + (size_t)(m0 + xrow) * (size_t)K + xc4;
    const float* wbase = Wm + (size_t)wr0 * (size_t)N + n0 + wc4;
    const size_t wstep16 = (size_t)16 * (size_t)N;

    v8f acc = {0.f,0.f,0.f,0.f,0.f,0.f,0.f,0.f};
    float4 xv, w0, w1;                        // in-flight staging registers

    auto gload = [&](int kk) {
        if (KAL) {
            xv = *(const float4*)(xbase + kk);
            w0 = *(const float4*)(wbase + (size_t)kk * (size_t)N);
            w1 = *(const float4*)(wbase + (size_t)kk * (size_t)N + wstep16);
        } else {
            float t0 = 0.f, t1 = 0.f, t2 = 0.f, t3 = 0.f;
            const int kc = kk + xc4;
            if (kc + 3 < K) {
                t0 = xbase[kk]; t1 = xbase[kk + 1];
                t2 = xbase[kk + 2]; t3 = xbase[kk + 3];
            } else {
                if (kc + 0 < K) t0 = xbase[kk];
                if (kc + 1 < K) t1 = xbase[kk + 1];
                if (kc + 2 < K) t2 = xbase[kk + 2];
                if (kc + 3 < K) t3 = xbase[kk + 3];
            }
            xv = make_float4(t0, t1, t2, t3);
            w0 = make_float4(0.f, 0.f, 0.f, 0.f);
            w1 = make_float4(0.f, 0.f, 0.f, 0.f);
            if (kk + wr0 < K)      w0 = *(const float4*)(wbase + (size_t)kk * (size_t)N);
            if (kk + wr0 + 16 < K) w1 = *(const float4*)(wbase + (size_t)kk * (size_t)N + wstep16);
        }
    };
    auto sstore = [&](int buf) {
        *(float4*)(&Xs[buf][xrow][xc4])     = xv;
        *(float4*)(&Ws[buf][wr0][wc4])      = w0;
        *(float4*)(&Ws[buf][wr0 + 16][wc4]) = w1;
    };

    const int nChunks = (K + 31) >> 5;
    gload(0);
    sstore(0);
    __syncthreads();

    int cur = 0;
    for (int ch = 0; ch < nChunks; ++ch) {
        const bool more = (ch + 1) < nChunks;
        if (more) gload((ch + 1) << 5);          // async: fills regs for next buf
        #pragma unroll
        for (int k4 = 0; k4 < 8; ++k4) {          // WMMAs on current buf (DS only)
            const int kb = (k4 << 2) + kb2;
            v2f a, b;
            a.x = Xs[cur][mrow][kb];
            a.y = Xs[cur][mrow][kb + 1];
            b.x = Ws[cur][kb][ncol];
            b.y = Ws[cur][kb + 1][ncol];
            acc = __builtin_amdgcn_wmma_f32_16x16x4_f32(
                      false, a, false, b, (short)0, acc, false, false);
        }
        if (more) sstore(cur ^ 1);                // waits loadcnt here, post-WMMA
        __syncthreads();
        cur ^= 1;
    }

    // epilogue: lane holds column (n0 + waveN*16 + l15), rows r + hi*8
    const int col = n0 + waveN * 16 + l15;
    const float bv = bias[col];
    const int rowBase = m0 + waveM * 16 + (hi << 3);
    #pragma unroll
    for (int r = 0; r < 8; ++r) {
        float v = acc[r] + bv;
        if (ACT == ACT_RELU)         v = v > 0.f ? v : 0.f;
        else if (ACT == ACT_SIGMOID) v = 1.0f / (1.0f + expf(-v));
        const size_t off = (size_t)(rowBase + r) * (size_t)N + col;
        if (ACCUM) Y[off] += v; else Y[off] = v;
    }
}

// ---------------------------------------------------------------------------
// LayerNorm: one 128-thread block per row; optional 3-way input sum and relu.
// ---------------------------------------------------------------------------
template<int NIN, bool RELU>
__global__ __launch_bounds__(128)
void ln_kernel(const float* __restrict__ x0, const float* __restrict__ x1,
               const float* __restrict__ x2, const float* __restrict__ g,
               const float* __restrict__ bb, float* __restrict__ y)
{
    __shared__ float red[128];
    const int e = blockIdx.x;
    const int t = threadIdx.x;
    float vals[3];
    float s = 0.f;
    #pragma unroll
    for (int j = 0; j < 3; ++j) {
        const int d = t + j * 128;
        const size_t off = (size_t)e * DIMV + d;
        float v = x0[off];
        if (NIN > 1) v += x1[off];
        if (NIN > 2) v += x2[off];
        vals[j] = v;
        s += v;
    }
    red[t] = s;
    __syncthreads();
    for (int st = 64; st > 0; st >>= 1) {
        if (t < st) red[t] += red[t + st];
        __syncthreads();
    }
    const float mean = red[0] * (1.0f / DIMV);
    __syncthreads();
    float vs = 0.f;
    #pragma unroll
    for (int j = 0; j < 3; ++j) { const float d = vals[j] - mean; vs += d * d; }
    red[t] = vs;
    __syncthreads();
    for (int st = 64; st > 0; st >>= 1) {
        if (t < st) red[t] += red[t + st];
        __syncthreads();
    }
    const float inv = rsqrtf(red[0] * (1.0f / DIMV) + 1e-3f);
    #pragma unroll
    for (int j = 0; j < 3; ++j) {
        const int d = t + j * 128;
        float o = (vals[j] - mean) * inv * g[d] + bb[d];
        if (RELU) o = o > 0.f ? o : 0.f;
        y[(size_t)e * DIMV + d] = o;
    }
}

// ---------------------------------------------------------------------------
// Row gather with -1 masking: dst[e,:] = idx[e] >= 0 ? src[idx[e],:] : 0
// ---------------------------------------------------------------------------
__global__ __launch_bounds__(256)
void gather_kernel(const float* __restrict__ src, const int* __restrict__ idx,
                   float* __restrict__ dst, int n4)
{
    const int t = blockIdx.x * 256 + threadIdx.x;
    if (t >= n4) return;
    const int e = t / (DIMV / 4);
    const int q = t % (DIMV / 4);
    const int id = idx[e];
    float4 v = make_float4(0.f, 0.f, 0.f, 0.f);
    if (id >= 0) v = ((const float4*)(src + (size_t)id * DIMV))[q];
    ((float4*)(dst + (size_t)e * DIMV))[q] = v;
}

// ---------------------------------------------------------------------------
// Segment softmax helpers (ordered-uint max trick + atomic adds)
// ---------------------------------------------------------------------------
__device__ __forceinline__ unsigned f2ord(float f) {
    unsigned u = __float_as_uint(f);
    return (u & 0x80000000u) ? ~u : (u | 0x80000000u);
}
__device__ __forceinline__ float ord2f(unsigned u) {
    return (u & 0x80000000u) ? __uint_as_float(u & 0x7FFFFFFFu)
                             : __uint_as_float(~u);
}

__global__ __launch_bounds__(256)
void segzero_kernel(unsigned* gm, float* es, float* y, int n)
{
    const int t = blockIdx.x * 256 + threadIdx.x;
    if (t >= n) return;
    gm[t] = 0u; es[t] = 0.f; y[t] = 0.f;
}

__global__ __launch_bounds__(256)
void segmax_kernel(const float* __restrict__ g, const int* __restrict__ seg,
                   unsigned* __restrict__ gmax, int n)
{
    const int t = blockIdx.x * 256 + threadIdx.x;
    if (t >= n) return;
    const int e = t / DIMV, d = t % DIMV;
    atomicMax(&gmax[(size_t)seg[e] * DIMV + d], f2ord(g[t]));
}

__global__ __launch_bounds__(256)
void segexp_kernel(float* __restrict__ g, const int* __restrict__ seg,
                   const unsigned* __restrict__ gmax, float* __restrict__ esum, int n)
{
    const int t = blockIdx.x * 256 + threadIdx.x;
    if (t >= n) return;
    const int e = t / DIMV, d = t % DIMV;
    const float v = expf(g[t] - ord2f(gmax[(size_t)seg[e] * DIMV + d]));
    g[t] = v;
    atomicAdd(&esum[(size_t)seg[e] * DIMV + d], v);
}

__global__ __launch_bounds__(256)
void segwsum_kernel(const float* __restrict__ f, const float* __restrict__ ev,
                    const int* __restrict__ seg, const float* __restrict__ esum,
                    float* __restrict__ y, int n)
{
    const int t = blockIdx.x * 256 + threadIdx.x;
    if (t >= n) return;
    const int e = t / DIMV, d = t % DIMV;
    const size_t so = (size_t)seg[e] * DIMV + d;
    atomicAdd(&y[so], f[t] * ev[t] / esum[so]);
}

__global__ __launch_bounds__(256)
void addseg_kernel(float* __restrict__ net, const float* __restrict__ z,
                   const int* __restrict__ seg, int n)
{
    const int t = blockIdx.x * 256 + threadIdx.x;
    if (t >= n) return;
    const int e = t / DIMV, d = t % DIMV;
    net[t] += z[(size_t)seg[e] * DIMV + d];
}

// out = base + s * m   (elementwise, float4)
__global__ __launch_bounds__(256)
void gated_kernel(float* __restrict__ outp, const float* __restrict__ base,
                  const float* __restrict__ s, const float* __restrict__ m, int n4)
{
    const int t = blockIdx.x * 256 + threadIdx.x;
    if (t >= n4) return;
    const float4 b4 = ((const float4*)base)[t];
    const float4 s4 = ((const float4*)s)[t];
    const float4 m4 = ((const float4*)m)[t];
    float4 o;
    o.x = b4.x + s4.x * m4.x;  o.y = b4.y + s4.y * m4.y;
    o.z = b4.z + s4.z * m4.z;  o.w = b4.w + s4.w * m4.w;
    ((float4*)outp)[t] = o;
}

// Output heads: one wave per edge; r = relu(net); d = r@Wout0+b; w = sigmoid(r@Wout1+b)
__global__ __launch_bounds__(256)
void heads_kernel(const float* __restrict__ net, const float* __restrict__ wout,
                  const float* __restrict__ bout, float* __restrict__ dout,
                  float* __restrict__ wdst)
{
    const int wid  = (blockIdx.x * 256 + threadIdx.x) >> 5;   // edge id
    const int lane = threadIdx.x & 31;
    if (wid >= EDGES) return;
    const float* row = net + (size_t)wid * DIMV;
    float a0 = 0.f, a1 = 0.f, a2 = 0.f, a3 = 0.f;
    for (int k = lane; k < DIMV; k += 32) {
        float r = row[k]; r = r > 0.f ? r : 0.f;
        a0 += r * wout[k * 2 + 0];
        a1 += r * wout[k * 2 + 1];
        a2 += r * wout[768 + k * 2 + 0];
        a3 += r * wout[768 + k * 2 + 1];
    }
    for (int m = 16; m; m >>= 1) {
        a0 += __shfl_xor(a0, m, 32);
        a1 += __shfl_xor(a1, m, 32);
        a2 += __shfl_xor(a2, m, 32);
        a3 += __shfl_xor(a3, m, 32);
    }
    if (lane == 0) {
        dout[wid * 2 + 0] = a0 + bout[0];
        dout[wid * 2 + 1] = a1 + bout[1];
        wdst[wid * 2 + 0] = 1.0f / (1.0f + expf(-(a2 + bout[2])));
        wdst[wid * 2 + 1] = 1.0f / (1.0f + expf(-(a3 + bout[3])));
    }
}

// ---------------------------------------------------------------------------
extern "C" void kernel_launch(void* const* d_in, const int* in_sizes, int n_in,
                              void* d_out, int out_size, void* d_ws, size_t ws_size,
                              hipStream_t stream)
{
    (void)in_sizes; (void)n_in; (void)out_size; (void)ws_size;

    const float* in_net = (const float*)d_in[0];
    const float* in_inp = (const float*)d_in[1];
    const float* corr   = (const float*)d_in[2];
    const int*   ix     = (const int*)d_in[3];
    const int*   jx     = (const int*)d_in[4];
    const int*   kkv    = (const int*)d_in[5];
    const int*   ijv    = (const int*)d_in[6];
    const float* Wc0    = (const float*)d_in[7];
    const float* bc0    = (const float*)d_in[8];
    const float* W      = (const float*)d_in[9];
    const float* b      = (const float*)d_in[10];
    const float* Wout   = (const float*)d_in[11];
    const float* bout   = (const float*)d_in[12];
    const float* lng    = (const float*)d_in[13];
    const float* lnb    = (const float*)d_in[14];

    float* out  = (float*)d_out;
    float* NET  = out;                               // [E,384] = final net output
    float* outD = out + (size_t)EDGES * DIMV;        // [E,2]
    float* outW = outD + (size_t)EDGES * 2;          // [E,2]

    const size_t ED = (size_t)EDGES * DIMV;
    const size_t SD = (size_t)NSEG * DIMV;
    float* ws = (float*)d_ws;
    float* T0  = ws;
    float* T1  = ws + ED;
    float* T2  = ws + 2 * ED;
    float* SGM = ws + 3 * ED;       // gmax (as uint) / reused as Z
    float* SES = SGM + SD;          // esum
    float* SY  = SES + SD;          // weighted sum y

    const int nED   = (int)ED;                 // 25,165,824
    const int nED4  = (int)(ED / 4);
    const int nSD   = (int)SD;
    const dim3 blk(256);
    const dim3 gE(EDGES / 32, DIMV / 64);      // 2048 x 6
    const dim3 gS(NSEG / 32, DIMV / 64);       // 128 x 6
    const int gEW  = (nED + 255) / 256;
    const int gEW4 = (nED4 + 255) / 256;
    const int gSW  = (nSD + 255) / 256;

    auto Wl = [&](int i) { return W + (size_t)i * DIMV * DIMV; };
    auto bl = [&](int i) { return b + (size_t)i * DIMV; };

    // ---- corr encoder: Linear-ReLU-Linear-LN-ReLU-Linear ----
    gemm_wmma<ACT_RELU, false, false><<<gE, blk, 0, stream>>>(
        corr, Wc0, bc0, T0, EDGES, CDIMK, DIMV);
    gemm_wmma<ACT_NONE, false, true><<<gE, blk, 0, stream>>>(T0, Wl(0), bl(0), T1, EDGES, DIMV, DIMV);
    ln_kernel<1, true><<<EDGES, 128, 0, stream>>>(T1, nullptr, nullptr, lng + 0, lnb + 0, T0);
    gemm_wmma<ACT_NONE, false, true><<<gE, blk, 0, stream>>>(T0, Wl(1), bl(1), T1, EDGES, DIMV, DIMV);

    // ---- net = LN(net + inp + c) ----
    ln_kernel<3, false><<<EDGES, 128, 0, stream>>>(in_net, in_inp, T1,
                                                   lng + DIMV, lnb + DIMV, NET);

    // ---- neighbor message passing (ix then jx) ----
    gather_kernel<<<gEW4, blk, 0, stream>>>(NET, ix, T0, nED4);
    gemm_wmma<ACT_RELU, false, true><<<gE, blk, 0, stream>>>(T0, Wl(2), bl(2), T1, EDGES, DIMV, DIMV);
    gemm_wmma<ACT_NONE, true,  true><<<gE, blk, 0, stream>>>(T1, Wl(3), bl(3), NET, EDGES, DIMV, DIMV);

    gather_kernel<<<gEW4, blk, 0, stream>>>(NET, jx, T0, nED4);
    gemm_wmma<ACT_RELU, false, true><<<gE, blk, 0, stream>>>(T0, Wl(4), bl(4), T1, EDGES, DIMV, DIMV);
    gemm_wmma<ACT_NONE, true,  true><<<gE, blk, 0, stream>>>(T1, Wl(5), bl(5), NET, EDGES, DIMV, DIMV);

    // ---- SoftAgg (x2) ----
    auto soft_agg = [&](const int* seg, const float* Wf, const float* bf,
                        const float* Wg, const float* bg,
                        const float* Wh, const float* bh) {
        gemm_wmma<ACT_NONE, false, true><<<gE, blk, 0, stream>>>(NET, Wg, bg, T0, EDGES, DIMV, DIMV); // g
        gemm_wmma<ACT_NONE, false, true><<<gE, blk, 0, stream>>>(NET, Wf, bf, T1, EDGES, DIMV, DIMV); // f
        segzero_kernel<<<gSW, blk, 0, stream>>>((unsigned*)SGM, SES, SY, nSD);
        segmax_kernel <<<gEW, blk, 0, stream>>>(T0, seg, (unsigned*)SGM, nED);
        segexp_kernel <<<gEW, blk, 0, stream>>>(T0, seg, (const unsigned*)SGM, SES, nED);
        segwsum_kernel<<<gEW, blk, 0, stream>>>(T1, T0, seg, SES, SY, nED);
        gemm_wmma<ACT_NONE, false, true><<<gS, blk, 0, stream>>>(SY, Wh, bh, SGM, NSEG, DIMV, DIMV);  // Z
        addseg_kernel <<<gEW, blk, 0, stream>>>(NET, SGM, seg, nED);
    };
    soft_agg(kkv, Wl(6), bl(6), Wl(7),  bl(7),  Wl(8),  bl(8));
    soft_agg(ijv, Wl(9), bl(9), Wl(10), bl(10), Wl(11), bl(11));

    // ---- GRU: LN -> GatedResidual -> LN -> GatedResidual ----
    ln_kernel<1, false><<<EDGES, 128, 0, stream>>>(NET, nullptr, nullptr,
                                                   lng + 2 * DIMV, lnb + 2 * DIMV, T0); // x
    gemm_wmma<ACT_SIGMOID, false, true><<<gE, blk, 0, stream>>>(T0, Wl(12), bl(12), T1, EDGES, DIMV, DIMV);
    gemm_wmma<ACT_RELU,    false, true><<<gE, blk, 0, stream>>>(T0, Wl(13), bl(13), T2, EDGES, DIMV, DIMV);
    gemm_wmma<ACT_NONE,    false, true><<<gE, blk, 0, stream>>>(T2, Wl(14), bl(14), NET, EDGES, DIMV, DIMV);
    gated_kernel<<<gEW4, blk, 0, stream>>>(T0, T0, T1, NET, nED4);   // x += sig * mlp

    ln_kernel<1, false><<<EDGES, 128, 0, stream>>>(T0, nullptr, nullptr,
                                                   lng + 3 * DIMV, lnb + 3 * DIMV, T1); // x2
    gemm_wmma<ACT_SIGMOID, false, true><<<gE, blk, 0, stream>>>(T1, Wl(15), bl(15), T0, EDGES, DIMV, DIMV);
    gemm_wmma<ACT_RELU,    false, true><<<gE, blk, 0, stream>>>(T1, Wl(16), bl(16), T2, EDGES, DIMV, DIMV);
    gemm_wmma<ACT_NONE,    false, true><<<gE, blk, 0, stream>>>(T2, Wl(17), bl(17), NET, EDGES, DIMV, DIMV);
    gated_kernel<<<gEW4, blk, 0, stream>>>(NET, T1, T0, NET, nED4);  // net = x2 + sig * mlp

    // ---- output heads ----
    heads_kernel<<<EDGES / 8, blk, 0, stream>>>(NET, Wout, bout, outD, outW);
}